// OverlapingLayer_45372034515105
// MI455X (gfx1250) — compile-verified
//
#include <hip/hip_runtime.h>

// out[b,i,h,j] = w1*x[b,h,j] + w2*(bias - x[b,h, (j<i ? j : j-i)])
// B=4, H=W=256. Output 256MB write-streaming -> store-bandwidth bound.
// CDNA5 paths used: global_load_async_to_lds_b128 (ASYNCcnt DMA), s_wait_asynccnt,
// non-temporal global_store_b128.

typedef __attribute__((ext_vector_type(4))) float v4f;

__device__ __forceinline__ unsigned umin_u32(unsigned a, unsigned b) {
    return a < b ? a : b;
}

__global__ __launch_bounds__(256) void overlap_layer_kernel(
    const float* __restrict__ x,
    const float* __restrict__ pb,
    const float* __restrict__ pw1,
    const float* __restrict__ pw2,
    float* __restrict__ out)
{
    __shared__ float row[256];

    const int tid = threadIdx.x;
    const int bh  = blockIdx.x;      // b*256 + h
    const int b   = bh >> 8;
    const int h   = bh & 255;

    // ---- async DMA: copy x[b,h,0:256] (1 KB) into LDS ----
    // Low 32 bits of a flat LDS pointer == LDS byte offset (aperture in high bits).
    const unsigned lds_base = (unsigned)(size_t)(void*)row;
    if (tid < 64) {
        const float* gsrc = x + ((size_t)bh << 8) + (size_t)(tid << 2); // 16B/lane
        const unsigned ldst = lds_base + (unsigned)(tid << 4);
        asm volatile("global_load_async_to_lds_b128 %0, %1, off"
                     :: "v"(ldst), "v"(gsrc)
                     : "memory");
    }
    asm volatile("s_wait_asynccnt 0" ::: "memory");
    __syncthreads();

    const float bias = pb[0];
    const float w1   = pw1[0];
    const float w2   = pw2[0];
    const float cb   = w2 * bias;   // hoisted constant
    const float nw2  = -w2;

    const int j0 = (tid & 63) << 2;   // 0..252 step 4 (float4-aligned column)
    const int i0 = tid >> 6;          // 0..3

    // t_e = w1*x[b,h,j] + w2*bias  (invariant over i)
    const float t0 = fmaf(w1, row[j0 + 0], cb);
    const float t1 = fmaf(w1, row[j0 + 1], cb);
    const float t2 = fmaf(w1, row[j0 + 2], cb);
    const float t3 = fmaf(w1, row[j0 + 3], cb);

    // flat index: b*2^24 + i*2^16 + h*2^8 + j
    float* op = out + (((size_t)b << 24) | ((size_t)i0 << 16) |
                       ((size_t)h << 8) | (size_t)j0);

#pragma unroll 4
    for (int k = 0; k < 64; ++k) {
        const int i = i0 + (k << 2);
        // idx(i,j) = min_u32(j, j-i): j-i wraps to huge when j<i -> picks j.
        const unsigned jm = (unsigned)(j0 - i);
        const float g0 = row[umin_u32((unsigned)(j0 + 0), jm + 0u)];
        const float g1 = row[umin_u32((unsigned)(j0 + 1), jm + 1u)];
        const float g2 = row[umin_u32((unsigned)(j0 + 2), jm + 2u)];
        const float g3 = row[umin_u32((unsigned)(j0 + 3), jm + 3u)];

        v4f v;
        v.x = fmaf(nw2, g0, t0);
        v.y = fmaf(nw2, g1, t1);
        v.z = fmaf(nw2, g2, t2);
        v.w = fmaf(nw2, g3, t3);

        // write-once 256MB stream: non-temporal 128-bit store
        __builtin_nontemporal_store(v, (v4f*)op);
        op += (size_t)4 << 16;   // i += 4
    }
}

extern "C" void kernel_launch(void* const* d_in, const int* in_sizes, int n_in,
                              void* d_out, int out_size, void* d_ws, size_t ws_size,
                              hipStream_t stream) {
    const float* x  = (const float*)d_in[0];
    const float* bb = (const float*)d_in[1];
    const float* w1 = (const float*)d_in[2];
    const float* w2 = (const float*)d_in[3];
    float* out = (float*)d_out;

    // one block per (b, h) row: B*H = in_sizes[0] / W  (W = 256)
    const int nblocks = in_sizes[0] >> 8;   // 4*256 = 1024
    overlap_layer_kernel<<<nblocks, 256, 0, stream>>>(x, bb, w1, w2, out);
}